// GraphSAGE_541165879479
// MI455X (gfx1250) — compile-verified
//
#include <hip/hip_runtime.h>

#define N_NODES 100000
#define N_EDGES 1600000
#define D 128
#define LDS_STRIDE 132   // pad 128 -> 132 floats: lane banks spread across all 64 banks

typedef __attribute__((ext_vector_type(2))) float v2f;
typedef __attribute__((ext_vector_type(8))) float v8f;

// ---------------- utility kernels ----------------

__global__ void zero_f32(float* __restrict__ p, int n) {
    int i = blockIdx.x * blockDim.x + threadIdx.x;
    if (i < n) p[i] = 0.0f;
}

__global__ void degree_kernel(const long long* __restrict__ dstIdx,
                              float* __restrict__ deg, int E) {
    int e = blockIdx.x * blockDim.x + threadIdx.x;
    if (e < E) atomicAdd(&deg[(int)dstIdx[e]], 1.0f);
}

__global__ void rdeg_kernel(float* __restrict__ deg, int n) {
    int i = blockIdx.x * blockDim.x + threadIdx.x;
    if (i < n) deg[i] = 1.0f / fmaxf(deg[i], 1.0f);
}

// One wave per edge; lane l moves feature chunk [4l, 4l+4) with f32 atomics.
// agg (51 MB) fits in the 192 MB L2, so these are L2-resident RMW atomics.
__global__ void scatter_kernel(const float* __restrict__ feat,
                               const long long* __restrict__ srcIdx,
                               const long long* __restrict__ dstIdx,
                               float* __restrict__ agg, int E) {
    long gid = (long)blockIdx.x * blockDim.x + threadIdx.x;
    int e    = (int)(gid >> 5);
    int lane = (int)(gid & 31);
    if (e >= E) return;
    int s = (int)srcIdx[e];
    int d = (int)dstIdx[e];
    const float4 v = *(const float4*)(feat + (long)s * D + lane * 4);
    float* dp = agg + (long)d * D + lane * 4;
    atomicAdd(dp + 0, v.x);
    atomicAdd(dp + 1, v.y);
    atomicAdd(dp + 2, v.z);
    atomicAdd(dp + 3, v.w);
}

// ---------------- fused SAGE linear: out = relu?(mean@Wl + b + x@Wr) ----------------
// Block = 256 threads = 8 waves. Block owns rows [16*bx, 16*bx+16).
// Wave w owns output columns [16w, 16w+16). f32 WMMA 16x16x4, K-loop of 32.

__global__ __launch_bounds__(256) void sage_linear_wmma(
    const float* __restrict__ agg,   // [N,128] summed neighbor features
    const float* __restrict__ rdeg,  // [N]     1/max(deg,1)
    const float* __restrict__ x,     // [N,128] self features
    const float* __restrict__ Wl,    // [128,128] row-major
    const float* __restrict__ Wr,    // [128,128] row-major
    const float* __restrict__ bias,  // [128]
    float* __restrict__ out,         // [N,128]
    int relu)
{
    __shared__ float As[16 * LDS_STRIDE];  // mean tile (pre-scaled by rdeg)
    __shared__ float Xs[16 * LDS_STRIDE];  // x tile

    const int rowBase = blockIdx.x * 16;   // N divisible by 16 -> no guards
    const int tid = threadIdx.x;

    // Stage tiles: 16 rows x 32 float4 chunks = 512 chunks, 2 per thread.
    #pragma unroll
    for (int it = 0; it < 2; ++it) {
        int id = tid + it * 256;           // 0..511
        int r  = id >> 5;                  // 0..15
        int c4 = id & 31;                  // 0..31
        long g = (long)(rowBase + r) * D + c4 * 4;
        float rd = rdeg[rowBase + r];
        float4 a = *(const float4*)(agg + g);
        a.x *= rd; a.y *= rd; a.z *= rd; a.w *= rd;
        *(float4*)(&As[r * LDS_STRIDE + c4 * 4]) = a;
        *(float4*)(&Xs[r * LDS_STRIDE + c4 * 4]) = *(const float4*)(x + g);
    }
    __syncthreads();

    const int wave    = tid >> 5;          // 0..7 -> output column tile
    const int lane    = tid & 31;
    const int nloc    = lane & 15;         // A: row M; B/D: column N
    const int khalf   = lane >> 4;         // 0: K=0,1   1: K=2,3
    const int colBase = wave * 16;

    v8f c = {};                            // f32 accumulator, 8 VGPRs
    #pragma unroll 4
    for (int kb = 0; kb < 32; ++kb) {
        const int k0 = kb * 4 + khalf * 2;
        // A fragments (16x4): lane holds A[nloc][k0], A[nloc][k0+1]
        v2f am = *(const v2f*)(&As[nloc * LDS_STRIDE + k0]);
        v2f ax = *(const v2f*)(&Xs[nloc * LDS_STRIDE + k0]);
        // B fragments (4x16): lane holds W[k0][col], W[k0+1][col]
        const float* wlp = Wl + (long)k0 * D + colBase + nloc;
        const float* wrp = Wr + (long)k0 * D + colBase + nloc;
        v2f bl = { wlp[0], wlp[D] };
        v2f br = { wrp[0], wrp[D] };
        // D = A*B + C   (8 args: neg_a, A, neg_b, B, c_mod, C, reuse_a, reuse_b)
        c = __builtin_amdgcn_wmma_f32_16x16x4_f32(false, am, false, bl, (short)0, c, false, false);
        c = __builtin_amdgcn_wmma_f32_16x16x4_f32(false, ax, false, br, (short)0, c, false, false);
    }

    const float bn = bias[colBase + nloc];
    #pragma unroll
    for (int v = 0; v < 8; ++v) {
        float o = c[v] + bn;
        if (relu) o = fmaxf(o, 0.0f);
        // D layout: VGPR v holds row m = v + 8*khalf, column nloc
        out[(long)(rowBase + v + 8 * khalf) * D + colBase + nloc] = o;
    }
}

// ---------------- launch ----------------

extern "C" void kernel_launch(void* const* d_in, const int* in_sizes, int n_in,
                              void* d_out, int out_size, void* d_ws, size_t ws_size,
                              hipStream_t stream) {
    const float*     x   = (const float*)d_in[0];
    const long long* ei  = (const long long*)d_in[1];   // int64 [2, E]
    const float*     W1l = (const float*)d_in[2];
    const float*     b1  = (const float*)d_in[3];
    const float*     W1r = (const float*)d_in[4];
    const float*     W2l = (const float*)d_in[5];
    const float*     b2  = (const float*)d_in[6];
    const float*     W2r = (const float*)d_in[7];
    float* out = (float*)d_out;

    const long long* srcIdx = ei;
    const long long* dstIdx = ei + N_EDGES;

    // workspace layout: agg [N*128] | deg/rdeg [N] | h [N*128]
    float* agg = (float*)d_ws;
    float* deg = agg + (size_t)N_NODES * D;
    float* h   = deg + N_NODES;

    const int ZB  = 256;
    const int nAgg = N_NODES * D;
    dim3 zgridA((nAgg + ZB - 1) / ZB), zgridN((N_NODES + ZB - 1) / ZB);
    dim3 egrid((N_EDGES + ZB - 1) / ZB);
    dim3 sgrid((N_EDGES * 32 + ZB - 1) / ZB);
    dim3 ggrid(N_NODES / 16);          // 6250 row tiles, exact
    dim3 blk(ZB);

    // degrees (shared by both layers) + zero accumulator
    zero_f32<<<zgridA, blk, 0, stream>>>(agg, nAgg);
    zero_f32<<<zgridN, blk, 0, stream>>>(deg, N_NODES);
    degree_kernel<<<egrid, blk, 0, stream>>>(dstIdx, deg, N_EDGES);
    rdeg_kernel<<<zgridN, blk, 0, stream>>>(deg, N_NODES);   // deg := 1/max(deg,1)

    // layer 1: aggregate x, then h = relu(mean@W1l + b1 + x@W1r)
    scatter_kernel<<<sgrid, blk, 0, stream>>>(x, srcIdx, dstIdx, agg, N_EDGES);
    sage_linear_wmma<<<ggrid, blk, 0, stream>>>(agg, deg, x, W1l, W1r, b1, h, 1);

    // layer 2: aggregate h, then out = mean@W2l + b2 + h@W2r
    zero_f32<<<zgridA, blk, 0, stream>>>(agg, nAgg);
    scatter_kernel<<<sgrid, blk, 0, stream>>>(h, srcIdx, dstIdx, agg, N_EDGES);
    sage_linear_wmma<<<ggrid, blk, 0, stream>>>(agg, deg, h, W2l, W2r, b2, out, 0);
}